// Custom_Dropout_51883204935704
// MI455X (gfx1250) — compile-verified
//
#include <hip/hip_runtime.h>

// Random-erase / cutout: out = (pixel covered by any of 8 boxes) ? 0 : x
// x: [B=32, C=3, W=512, H=512] fp32, H contiguous.
// width_positions / height_positions: [B, C, 8, 2] int32 ({start,end}, end-exclusive).
//
// MI455X strategy (compile-only reasoning):
//  * 201 MB traffic @ 23.3 TB/s ~ 8.6 us -> HBM-bound; the binding secondary
//    constraint is instruction issue, so amortize per-wave overhead:
//    1 block = 4096 elems (8 rows); each thread owns 4 float4s at stride 1024.
//  * h = (4*tid) & 511 is identical for all 4 sub-tiles -> per-box 4-bit
//    H-coverage mask computed ONCE per thread, reused for 4 rows.
//  * Wave32 geometry: each wave handles rows {r, r+2, r+4, r+6}, r wave-uniform
//    (readfirstlane) -> all W-interval tests and box-skip guards are pure SALU.
//  * Descriptors: 2x s_load_b512 up front, one KMcnt drain (round-1 lesson).
//  * 192MB L2: temporal b128 loads keep x resident across graph replays;
//    non-temporal b128 stores keep the write stream from evicting it.

typedef float v4f __attribute__((ext_vector_type(4)));
typedef int   v4i __attribute__((ext_vector_type(4)));

#define HDIM 512
#define WDIM 512
#define PLANE_ELEMS (WDIM * HDIM)     // 262144
#define NBOX 8
#define BLOCK_THREADS 256
#define ELEMS_PER_BLOCK 4096          // 8 rows; 4 float4 per thread
#define BLOCKS_PER_PLANE (PLANE_ELEMS / ELEMS_PER_BLOCK)  // 64

__global__ __launch_bounds__(BLOCK_THREADS) void cutout_kernel(
    const float* __restrict__ x,
    const int*   __restrict__ wpos,   // [planes, NBOX, 2]
    const int*   __restrict__ hpos,   // [planes, NBOX, 2]
    float*       __restrict__ out)
{
    const int plane = (int)blockIdx.x >> 6;          // BLOCKS_PER_PLANE == 64
    const int blk   = (int)blockIdx.x & 63;
    const int tid   = (int)threadIdx.x;

    const int h       = (tid * 4) & (HDIM - 1);      // same for all 4 sub-tiles
    const int halfsel = __builtin_amdgcn_readfirstlane(tid >> 7);  // 0/1, wave-uniform
    const int r       = blk * 8 + halfsel;           // wave's rows: r, r+2, r+4, r+6

    const int idx0 = plane * (PLANE_ELEMS / 4) + blk * (ELEMS_PER_BLOCK / 4) + tid;

    // Descriptor loads first: uniform int4 -> s_load_b128 (fused to s_load_b512),
    // single KMcnt drain, overlapped with the bulk loads below.
    const v4i* __restrict__ wp4 = (const v4i*)(wpos + plane * (NBOX * 2));
    const v4i* __restrict__ hp4 = (const v4i*)(hpos + plane * (NBOX * 2));
    const v4i w01 = wp4[0], w23 = wp4[1], w45 = wp4[2], w67 = wp4[3];
    const v4i h01 = hp4[0], h23 = hp4[1], h45 = hp4[2], h67 = hp4[3];

    // 4 coalesced b128 loads in flight per wave (MLP for latency hiding).
    const v4f* __restrict__ x4 = (const v4f*)x;
    v4f v[4];
    v[0] = x4[idx0];
    v[1] = x4[idx0 + 256];
    v[2] = x4[idx0 + 512];
    v[3] = x4[idx0 + 768];

    const int ws[NBOX] = { w01.x, w01.z, w23.x, w23.z, w45.x, w45.z, w67.x, w67.z };
    const int we[NBOX] = { w01.y, w01.w, w23.y, w23.w, w45.y, w45.w, w67.y, w67.w };
    const int hs[NBOX] = { h01.x, h01.z, h23.x, h23.z, h45.x, h45.z, h67.x, h67.z };
    const int he[NBOX] = { h01.y, h01.w, h23.y, h23.w, h45.y, h45.w, h67.y, h67.w };

    // Per-box 4-bit H coverage of this thread's 4 columns: computed once,
    // reused for all 4 rows. Scalar guard skips boxes nowhere near rows r..r+6.
    unsigned m[NBOX];
#pragma unroll
    for (int n = 0; n < NBOX; ++n) {
        m[n] = 0u;
        if (ws[n] <= r + 6 && we[n] > r) {           // SALU guard, wave-coherent
            const unsigned span = (unsigned)(he[n] - hs[n]);
            const int      t    = h - hs[n];
            m[n] = ((unsigned)(t + 0) < span ? 1u : 0u)
                 | ((unsigned)(t + 1) < span ? 2u : 0u)
                 | ((unsigned)(t + 2) < span ? 4u : 0u)
                 | ((unsigned)(t + 3) < span ? 8u : 0u);
        }
    }

#pragma unroll
    for (int j = 0; j < 4; ++j) {
        const int wj = r + 2 * j;                    // scalar row index
        unsigned cov = 0u;
#pragma unroll
        for (int n = 0; n < NBOX; ++n) {
            // Pure-SALU interval test + branch; body is a single v_or.
            if (wj >= ws[n] && wj < we[n]) cov |= m[n];
        }
        v4f vv = v[j];
        if (cov & 1u) vv.x = 0.0f;
        if (cov & 2u) vv.y = 0.0f;
        if (cov & 4u) vv.z = 0.0f;
        if (cov & 8u) vv.w = 0.0f;
        // Write-once output: non-temporal b128 store (th:NT).
        __builtin_nontemporal_store(vv, (v4f*)out + idx0 + j * 256);
    }
}

extern "C" void kernel_launch(void* const* d_in, const int* in_sizes, int n_in,
                              void* d_out, int out_size, void* d_ws, size_t ws_size,
                              hipStream_t stream) {
    const float* x   = (const float*)d_in[0];  // [B,C,512,512] fp32
    const int*   wp  = (const int*)d_in[1];    // [B,C,8,2] int32 (W ranges)
    const int*   hp  = (const int*)d_in[2];    // [B,C,8,2] int32 (H ranges)
    float*       out = (float*)d_out;

    const int planes  = in_sizes[0] / PLANE_ELEMS;   // 32*3 = 96
    const int nblocks = planes * BLOCKS_PER_PLANE;   // 6144

    cutout_kernel<<<dim3(nblocks), dim3(BLOCK_THREADS), 0, stream>>>(x, wp, hp, out);
}